// sanchitLoss_23914377904445
// MI455X (gfx1250) — compile-verified
//
#include <hip/hip_runtime.h>

// sanchitLoss: mean over B rows of sum_c dis(true,c) * softmax(pred)_c
// B = 8388608, C = 4. Memory-bound streaming reduction:
//   reads 134MB pred (float4/row) + 33.5MB true(int32) -> ~7.2us floor @23.3TB/s.
// Whole working set (168MB) fits the 192MB L2, so replayed timing runs L2-resident.
// Matrix pipe is idle, so wave-level reduction runs on V_WMMA_F32_16X16X4_F32.

typedef __attribute__((ext_vector_type(2))) float v2f;
typedef __attribute__((ext_vector_type(8))) float v8f;

#define BLOCK 256
#define GRID  4096   // 1M threads, 8 rows/thread at B=8388608

// Wave32 reduction through the matrix pipe:
// A = lane partials as a 16x4 tile (lane L<16 -> A[M=L,K=0], lane L>=16 ->
// A[M=L-16,K=2]; second VGPR zero), B = ones(4x16) -> D[m,n] = x_m + x_{m+16}.
// Lane 0 sums D rows 0..7 (lanes 0..7,16..23), lane 16 sums rows 8..15.
// Requires EXEC all-ones: only called from fully-active code points.
__device__ __forceinline__ float wave_reduce32(float x) {
  v2f a; a[0] = x;    a[1] = 0.0f;
  v2f b; b[0] = 1.0f; b[1] = 1.0f;
  v8f c = {};
  c = __builtin_amdgcn_wmma_f32_16x16x4_f32(
      /*neg_a=*/false, a, /*neg_b=*/false, b,
      /*c_mod=*/(short)0, c, /*reuse_a=*/false, /*reuse_b=*/false);
  float s = (c[0] + c[1]) + (c[2] + c[3]) + (c[4] + c[5]) + (c[6] + c[7]);
  return __shfl(s, 0, 32) + __shfl(s, 16, 32);
}

// Per-row loss: softmax-weighted expected distance.
// dis rows (closed form of the reference's masked fixups):
//   t=0:[0,1,2,3] t=1:[1,0,2,2] t=2:[2,2,0,1] t=3:[3,2,1,0]
__device__ __forceinline__ float row_loss(float4 q, int t) {
  const float m  = fmaxf(fmaxf(q.x, q.y), fmaxf(q.z, q.w));
  const float e0 = __expf(q.x - m);
  const float e1 = __expf(q.y - m);
  const float e2 = __expf(q.z - m);
  const float e3 = __expf(q.w - m);

  const float tf = (float)t;
  const float d0 = tf;
  const float d1 = fabsf(1.0f - tf) + (t == 2 ? 1.0f : 0.0f);
  const float d2 = fabsf(2.0f - tf) + (t == 1 ? 1.0f : 0.0f);
  const float d3 = 3.0f - tf;

  const float num = d0 * e0 + d1 * e1 + d2 * e2 + d3 * e3;
  const float den = (e0 + e1) + (e2 + e3);
  return num * __builtin_amdgcn_rcpf(den);
}

__device__ __forceinline__ void block_reduce_store(float acc, float* dst,
                                                   float scale) {
  const float wsum = wave_reduce32(acc);      // all lanes active here

  __shared__ float smem[BLOCK / 32];
  const int lane = threadIdx.x & 31;
  const int wid  = threadIdx.x >> 5;
  if (lane == 0) smem[wid] = wsum;
  __syncthreads();

  if (wid == 0) {
    float v = (lane < (BLOCK / 32)) ? smem[lane] : 0.0f;
    v += __shfl_down(v, 4);
    v += __shfl_down(v, 2);
    v += __shfl_down(v, 1);
    if (lane == 0) *dst = v * scale;          // deterministic: no atomics
  }
}

__global__ __launch_bounds__(BLOCK)
void sanchit_partial(const float4* __restrict__ pred,
                     const int*    __restrict__ tru,
                     float*        __restrict__ partial,
                     int nrows) {
  const int tid    = blockIdx.x * BLOCK + threadIdx.x;
  const int stride = gridDim.x * BLOCK;
  const int iters  = nrows / stride;          // uniform trip count, no per-iter cmp

  float acc = 0.0f;
  int r = tid;
  #pragma unroll 2
  for (int i = 0; i < iters; ++i, r += stride) {
    // Speculative prefetch of next stride: OOB translation silently dropped
    // (ISA 10.5); high locality keeps lines resident in the 192MB L2.
    __builtin_prefetch(pred + r + stride, 0, 3);   // global_prefetch_b8
    const float4 q = pred[r];                      // global_load_b128, coalesced
    const int    t = tru[r];                       // global_load_b32
    acc += row_loss(q, t);
  }
  if (r < nrows) {                                 // tail (empty when stride | B)
    acc += row_loss(pred[r], tru[r]);
  }

  block_reduce_store(acc, &partial[blockIdx.x], 1.0f);
}

__global__ __launch_bounds__(BLOCK)
void sanchit_final(const float* __restrict__ partial,
                   float*       __restrict__ out,
                   int nparts, float scale) {
  float acc = 0.0f;
  for (int i = threadIdx.x; i < nparts; i += BLOCK) acc += partial[i];
  block_reduce_store(acc, out, scale);
}

extern "C" void kernel_launch(void* const* d_in, const int* in_sizes, int n_in,
                              void* d_out, int out_size, void* d_ws, size_t ws_size,
                              hipStream_t stream) {
  const float4* pred = (const float4*)d_in[0];   // [B,4] f32
  const int*    tru  = (const int*)d_in[1];      // [B] int32 (JAX x64 off)
  const int     nrows = in_sizes[1];             // B

  float* partials = (float*)d_ws;                // GRID floats = 16KB scratch

  sanchit_partial<<<GRID, BLOCK, 0, stream>>>(pred, tru, partials, nrows);
  sanchit_final<<<1, BLOCK, 0, stream>>>(partials, (float*)d_out,
                                         GRID, 1.0f / (float)nrows);
}